// MAGNN_nc_mb_20624432955635
// MI455X (gfx1250) — compile-verified
//
#include <hip/hip_runtime.h>
#include <math.h>

// ---------------------------------------------------------------------------
// MAGNN_nc_mb forward for MI455X (gfx1250, wave32).
// Dense GEMM stages use V_WMMA_F32_16X16X4_F32 (full f32 precision).
// Gather/scatter stages sized so tf/ft/h/outs stay L2-resident (192 MB L2).
// ---------------------------------------------------------------------------

#define N_NODES 100000
#define N0      60000
#define N1      40000
#define F0      128
#define F1      64
#define HID     64
#define HEADS   8
#define OUTD    16
#define ATTNV   128
#define NE      250000
#define LP      3
#define NG      8192
#define NB      8192

typedef __attribute__((ext_vector_type(2))) float f32x2;
typedef __attribute__((ext_vector_type(8))) float f32x8;

__device__ __forceinline__ f32x8 wmma_f32(f32x2 a, f32x2 b, f32x8 c) {
  // D = A(16x4) * B(4x16) + C(16x16), all f32, wave32
  return __builtin_amdgcn_wmma_f32_16x16x4_f32(false, a, false, b, (short)0, c,
                                               false, false);
}

// ---------------------------------------------------------------------------
// generic fill
__global__ void fill_kernel(float* __restrict__ p, float v, int n) {
  int i = blockIdx.x * 256 + threadIdx.x;
  if (i < n) p[i] = v;
}

// ---------------------------------------------------------------------------
// RotatE suffix-product rotation constants for both metapaths.
// rv_full = [rv0, conj(rv0), rv1, conj(rv1)]; etypes mp0={0,1}, mp1={2,3}.
// finals[L-1]=1; finals[i]=cmul(finals[i+1], rv_full[etypes[i]]).
__global__ void finals_kernel(const float* __restrict__ r_vec,
                              float* __restrict__ finals /*[2][3][64]*/) {
  int c = threadIdx.x;
  if (c >= 32) return;
  for (int mp = 0; mp < 2; ++mp) {
    float re = r_vec[(mp * 32 + c) * 2 + 0];
    float im = r_vec[(mp * 32 + c) * 2 + 1];
    float inv = rsqrtf(re * re + im * im);
    re *= inv; im *= inv;
    // etypes[0] -> (re, im), etypes[1] -> (re, -im)
    float f1re = re, f1im = -im;                    // finals[1] = 1 * conj
    float f0re = f1re * re - f1im * im;             // finals[0] = finals[1]*e0
    float f0im = f1re * im + f1im * re;
    float* F = finals + mp * LP * HID;
    F[0 * HID + 2 * c] = f0re; F[0 * HID + 2 * c + 1] = f0im;
    F[1 * HID + 2 * c] = f1re; F[1 * HID + 2 * c + 1] = f1im;
    F[2 * HID + 2 * c] = 1.f;  F[2 * HID + 2 * c + 1] = 0.f;
  }
}

// ---------------------------------------------------------------------------
// Per-type linear: tf[idx[r]] = feat[r] @ W^T + b.  WMMA f32 16x16x4.
// Block = 8 waves; each wave owns a 16-row stripe, 4 N-tiles of 16 (HID=64).
// W^T staged to LDS as [K][64].
template <int K>
__global__ __launch_bounds__(256) void gemm_tf_kernel(
    const float* __restrict__ feat, const float* __restrict__ W,
    const float* __restrict__ bias, const int* __restrict__ idx,
    float* __restrict__ tf, int M) {
  __shared__ float ldsB[K * HID];
  int tid = threadIdx.x;
  for (int i = tid; i < K * HID; i += 256) {
    int k = i >> 6, n = i & 63;
    ldsB[i] = W[n * K + k];  // W is [HID][K] row-major -> transpose into LDS
  }
  __syncthreads();
  int wave = tid >> 5, lane = tid & 31;
  int tile_m = blockIdx.x * 128 + wave * 16;
  if (tile_m >= M) return;
  int half = lane >> 4, l15 = lane & 15;
  int arow = tile_m + l15;
  if (arow >= M) arow = M - 1;  // clamp; result rows >= M never stored
  f32x8 c0 = {}, c1 = {}, c2 = {}, c3 = {};
  for (int k0 = 0; k0 < K; k0 += 4) {
    int kk = k0 + half * 2;  // lanes 0-15: K=k0,k0+1 ; lanes 16-31: K=k0+2,k0+3
    f32x2 a;
    a.x = feat[(long)arow * K + kk];
    a.y = feat[(long)arow * K + kk + 1];
    f32x2 b0, b1, b2, b3;
    b0.x = ldsB[kk * HID + l15 + 0];  b0.y = ldsB[(kk + 1) * HID + l15 + 0];
    b1.x = ldsB[kk * HID + l15 + 16]; b1.y = ldsB[(kk + 1) * HID + l15 + 16];
    b2.x = ldsB[kk * HID + l15 + 32]; b2.y = ldsB[(kk + 1) * HID + l15 + 32];
    b3.x = ldsB[kk * HID + l15 + 48]; b3.y = ldsB[(kk + 1) * HID + l15 + 48];
    c0 = wmma_f32(a, b0, c0);
    c1 = wmma_f32(a, b1, c1);
    c2 = wmma_f32(a, b2, c2);
    c3 = wmma_f32(a, b3, c3);
  }
#pragma unroll
  for (int v = 0; v < 8; ++v) {
    int r = tile_m + v + half * 8;
    if (r < M) {
      float* drow = tf + (long)idx[r] * HID;
      drow[l15 + 0]  = c0[v] + bias[l15 + 0];
      drow[l15 + 16] = c1[v] + bias[l15 + 16];
      drow[l15 + 32] = c2[v] + bias[l15 + 32];
      drow[l15 + 48] = c3[v] + bias[l15 + 48];
    }
  }
}

// ---------------------------------------------------------------------------
// One wave per edge: gather 3 tf rows (L2-resident), rotate, mean -> h[e];
// attention logits a[e][8] via wave reduce; segment max via f32 atomicMax.
__global__ __launch_bounds__(256) void edge_h_kernel(
    const float* __restrict__ tf, const float* __restrict__ finals,
    const int* __restrict__ indices, const int* __restrict__ dst,
    const float* __restrict__ attn, float* __restrict__ h,
    float* __restrict__ a, float* __restrict__ m) {
  int wave = threadIdx.x >> 5, lane = threadIdx.x & 31;
  int e = blockIdx.x * 8 + wave;
  if (e >= NE) return;
  float2 acc = make_float2(0.f, 0.f);
#pragma unroll
  for (int i = 0; i < LP; ++i) {
    int node = indices[e * LP + i];
    const float* row = tf + (long)node * HID + lane * 2;
    float vx = row[0], vy = row[1];
    float fx = finals[i * HID + lane * 2], fy = finals[i * HID + lane * 2 + 1];
    acc.x += vx * fx - vy * fy;
    acc.y += vx * fy + vy * fx;
  }
  acc.x *= (1.f / 3.f);
  acc.y *= (1.f / 3.f);
  h[(long)e * HID + lane * 2] = acc.x;
  h[(long)e * HID + lane * 2 + 1] = acc.y;
  int d = dst[e];
#pragma unroll
  for (int hh = 0; hh < HEADS; ++hh) {
    float p = acc.x * attn[hh * HID + lane * 2] +
              acc.y * attn[hh * HID + lane * 2 + 1];
#pragma unroll
    for (int off = 16; off > 0; off >>= 1) p += __shfl_xor(p, off, 32);
    if (lane == 0) {
      float av = p > 0.f ? p : 0.01f * p;  // leaky_relu(0.01)
      a[(long)e * HEADS + hh] = av;
      atomicMax(&m[d * HEADS + hh], av);   // GLOBAL_ATOMIC_MAX_NUM_F32
    }
  }
}

// ex = exp(a - m[dst]); segment sum of ex
__global__ void edge_exp_kernel(float* __restrict__ a,
                                const int* __restrict__ dst,
                                const float* __restrict__ m,
                                float* __restrict__ s, int n) {
  int i = blockIdx.x * 256 + threadIdx.x;
  if (i >= n) return;
  int e = i >> 3, hh = i & 7;
  int d = dst[e];
  float ex = __expf(a[i] - m[d * HEADS + hh]);
  a[i] = ex;
  atomicAdd(&s[d * HEADS + hh], ex);
}

// ft[dst] += an * h  (one wave per edge, lane = one complex pair of HID)
__global__ __launch_bounds__(256) void edge_scatter_kernel(
    const float* __restrict__ h, const float* __restrict__ a,
    const float* __restrict__ s, const int* __restrict__ dst,
    float* __restrict__ ft) {
  int wave = threadIdx.x >> 5, lane = threadIdx.x & 31;
  int e = blockIdx.x * 8 + wave;
  if (e >= NE) return;
  int d = dst[e];
  float an_l = 0.f;
  if (lane < HEADS) an_l = a[(long)e * HEADS + lane] / s[d * HEADS + lane];
  float hx = h[(long)e * HID + lane * 2];
  float hy = h[(long)e * HID + lane * 2 + 1];
#pragma unroll
  for (int hh = 0; hh < HEADS; ++hh) {
    float w = __shfl(an_l, hh, 32);
    float* base = ft + ((long)d * HEADS + hh) * HID + lane * 2;
    atomicAdd(base + 0, w * hx);   // GLOBAL_ATOMIC_ADD_F32, L2-resident ft
    atomicAdd(base + 1, w * hy);
  }
}

// out_p[b] = elu(ft[target[b]])
__global__ void target_elu_kernel(const float* __restrict__ ft,
                                  const int* __restrict__ target,
                                  float* __restrict__ outp, int n) {
  int i = blockIdx.x * 256 + threadIdx.x;
  if (i >= n) return;
  int b = i >> 9, j = i & 511;
  float v = ft[(long)target[b] * (HEADS * HID) + j];
  outp[i] = v > 0.f ? v : (__expf(v) - 1.f);
}

// ---------------------------------------------------------------------------
// Fused metapath-attention score: sum_b tanh(out_p @ fc1_w^T + fc1_b) . fc2_w
// WMMA over [B,512] x [512,128]; per-lane tanh+dot, wave-reduced into score[p].
__global__ __launch_bounds__(256) void score_kernel(
    const float* __restrict__ outs, const float* __restrict__ fc1_w,
    const float* __restrict__ fc1_b, const float* __restrict__ fc2_w,
    float* __restrict__ score) {
  int p = blockIdx.y;
  const float* A = outs + (long)p * NB * (HEADS * HID);
  int wave = threadIdx.x >> 5, lane = threadIdx.x & 31;
  int tile_m = (blockIdx.x * 8 + wave) * 16;
  int half = lane >> 4, l15 = lane & 15;
  long arow = tile_m + l15;
  f32x8 acc[8] = {};
  for (int k0 = 0; k0 < HEADS * HID; k0 += 4) {
    int kk = k0 + half * 2;
    f32x2 af;
    af.x = A[arow * (HEADS * HID) + kk];
    af.y = A[arow * (HEADS * HID) + kk + 1];
#pragma unroll
    for (int nt = 0; nt < 8; ++nt) {
      int col = nt * 16 + l15;
      f32x2 bf;
      bf.x = fc1_w[(long)col * (HEADS * HID) + kk];      // fc1_w^T, L2-hot
      bf.y = fc1_w[(long)col * (HEADS * HID) + kk + 1];
      acc[nt] = wmma_f32(af, bf, acc[nt]);
    }
  }
  float ssum = 0.f;
#pragma unroll
  for (int nt = 0; nt < 8; ++nt) {
    int col = nt * 16 + l15;
    float bb = fc1_b[col], w2 = fc2_w[col];
#pragma unroll
    for (int v = 0; v < 8; ++v) ssum += tanhf(acc[nt][v] + bb) * w2;
  }
#pragma unroll
  for (int off = 16; off > 0; off >>= 1) ssum += __shfl_xor(ssum, off, 32);
  if (lane == 0) atomicAdd(&score[p], ssum);
}

__global__ void beta_kernel(const float* __restrict__ score,
                            float* __restrict__ beta) {
  float s0 = score[0] / (float)NB, s1 = score[1] / (float)NB;
  float mx = fmaxf(s0, s1);
  float e0 = __expf(s0 - mx), e1 = __expf(s1 - mx);
  float inv = 1.f / (e0 + e1);
  beta[0] = e0 * inv;
  beta[1] = e1 * inv;
}

__global__ void hagg_kernel(const float* __restrict__ outs,
                            const float* __restrict__ beta,
                            float* __restrict__ hagg, int n) {
  int i = blockIdx.x * 256 + threadIdx.x;
  if (i >= n) return;
  hagg[i] = beta[0] * outs[i] + beta[1] * outs[(long)n + i];
}

// logits = hagg @ fc_w^T + fc_b  ([8192,512] x [512,16]); fc_w^T staged to LDS
__global__ __launch_bounds__(256) void logits_kernel(
    const float* __restrict__ hagg, const float* __restrict__ fc_w,
    const float* __restrict__ fc_b, float* __restrict__ logits) {
  __shared__ float ldsB[(HEADS * HID) * OUTD];  // 32 KB
  int tid = threadIdx.x;
  for (int i = tid; i < (HEADS * HID) * OUTD; i += 256) {
    int k = i >> 4, n = i & 15;
    ldsB[i] = fc_w[n * (HEADS * HID) + k];
  }
  __syncthreads();
  int wave = tid >> 5, lane = tid & 31;
  int tile_m = (blockIdx.x * 8 + wave) * 16;
  int half = lane >> 4, l15 = lane & 15;
  long arow = tile_m + l15;
  f32x8 c = {};
  for (int k0 = 0; k0 < HEADS * HID; k0 += 4) {
    int kk = k0 + half * 2;
    f32x2 af;
    af.x = hagg[arow * (HEADS * HID) + kk];
    af.y = hagg[arow * (HEADS * HID) + kk + 1];
    f32x2 bf;
    bf.x = ldsB[kk * OUTD + l15];
    bf.y = ldsB[(kk + 1) * OUTD + l15];
    c = wmma_f32(af, bf, c);
  }
#pragma unroll
  for (int v = 0; v < 8; ++v) {
    int r = tile_m + v + half * 8;
    logits[(long)r * OUTD + l15] = c[v] + fc_b[l15];
  }
}

// ---------------------------------------------------------------------------
extern "C" void kernel_launch(void* const* d_in, const int* in_sizes, int n_in,
                              void* d_out, int out_size, void* d_ws,
                              size_t ws_size, hipStream_t stream) {
  const float* feat0 = (const float*)d_in[0];
  const float* feat1 = (const float*)d_in[1];
  const float* W0    = (const float*)d_in[2];
  const float* b0    = (const float*)d_in[3];
  const float* W1    = (const float*)d_in[4];
  const float* b1    = (const float*)d_in[5];
  const float* r_vec = (const float*)d_in[6];
  const float* attn0 = (const float*)d_in[7];
  const float* attn1 = (const float*)d_in[8];
  const float* fc1_w = (const float*)d_in[9];
  const float* fc1_b = (const float*)d_in[10];
  const float* fc2_w = (const float*)d_in[11];
  const float* fc_w  = (const float*)d_in[12];
  const float* fc_b  = (const float*)d_in[13];
  const int* idx0    = (const int*)d_in[14];
  const int* idx1    = (const int*)d_in[15];
  const int* mp_indices[2] = {(const int*)d_in[16], (const int*)d_in[19]};
  const int* mp_dst[2]     = {(const int*)d_in[17], (const int*)d_in[20]};
  const int* mp_target[2]  = {(const int*)d_in[18], (const int*)d_in[21]};
  const float* attn[2] = {attn0, attn1};

  // workspace layout (floats); total ~37.1M floats (~149 MB)
  float* ws     = (float*)d_ws;
  float* tf     = ws;                                   // N_NODES*HID = 6.4M
  float* finals = tf + (long)N_NODES * HID;             // 2*3*64
  float* h      = finals + 2 * LP * HID;                // NE*HID = 16M
  float* a      = h + (long)NE * HID;                   // NE*HEADS = 2M
  float* m      = a + (long)NE * HEADS;                 // NG*HEADS
  float* s      = m + NG * HEADS;                       // NG*HEADS
  float* ft     = s + NG * HEADS;                       // NG*HEADS*HID = 4.19M
  float* outs   = ft + (long)NG * HEADS * HID;          // 2*NB*512 = 8.39M
  float* score  = outs + 2L * NB * HEADS * HID;         // 2
  float* beta   = score + 2;                            // 2

  float* logits = (float*)d_out;                        // [NB, OUTD]
  float* hagg   = logits + (long)NB * OUTD;             // [NB, 512]

  // 1) per-type feature transform into dense tf table (L2-resident, 25.6 MB)
  gemm_tf_kernel<F0><<<(N0 + 127) / 128, 256, 0, stream>>>(feat0, W0, b0, idx0,
                                                           tf, N0);
  gemm_tf_kernel<F1><<<(N1 + 127) / 128, 256, 0, stream>>>(feat1, W1, b1, idx1,
                                                           tf, N1);
  // 2) rotation constants
  finals_kernel<<<1, 32, 0, stream>>>(r_vec, finals);

  // 3) per-metapath intra-path aggregation + node-level attention
  for (int mp = 0; mp < 2; ++mp) {
    fill_kernel<<<(NG * HEADS + 255) / 256, 256, 0, stream>>>(m, -3.0e38f,
                                                              NG * HEADS);
    fill_kernel<<<(NG * HEADS + 255) / 256, 256, 0, stream>>>(s, 0.f,
                                                              NG * HEADS);
    fill_kernel<<<(NG * HEADS * HID + 255) / 256, 256, 0, stream>>>(
        ft, 0.f, NG * HEADS * HID);
    edge_h_kernel<<<NE / 8, 256, 0, stream>>>(tf, finals + mp * LP * HID,
                                              mp_indices[mp], mp_dst[mp],
                                              attn[mp], h, a, m);
    edge_exp_kernel<<<(NE * HEADS + 255) / 256, 256, 0, stream>>>(
        a, mp_dst[mp], m, s, NE * HEADS);
    edge_scatter_kernel<<<NE / 8, 256, 0, stream>>>(h, a, s, mp_dst[mp], ft);
    target_elu_kernel<<<(NB * HEADS * HID + 255) / 256, 256, 0, stream>>>(
        ft, mp_target[mp], outs + (long)mp * NB * HEADS * HID,
        NB * HEADS * HID);
  }

  // 4) metapath-level attention
  fill_kernel<<<1, 64, 0, stream>>>(score, 0.f, 2);
  {
    dim3 g(NB / 128, 2);
    score_kernel<<<g, 256, 0, stream>>>(outs, fc1_w, fc1_b, fc2_w, score);
  }
  beta_kernel<<<1, 1, 0, stream>>>(score, beta);
  hagg_kernel<<<(NB * HEADS * HID + 255) / 256, 256, 0, stream>>>(
      outs, beta, hagg, NB * HEADS * HID);
  logits_kernel<<<NB / 128, 256, 0, stream>>>(hagg, fc_w, fc_b, logits);
}